// HLLUT_13477607375178
// MI455X (gfx1250) — compile-verified
//
#include <hip/hip_runtime.h>
#include <stdint.h>

// HLLUT 2x super-resolution, scatter-free formulation.
// img_lr: (4,1,512,512) int32 in [0,256)
// h_weight/l_weight: (256^3, 4) float32  (256MB each)
// out: (4,1,1024,1024) float32
//
// Per LR pixel (y,x), the 8 reference passes (h/l kernel x 4 rotations)
// reduce to 8 LUT gathers whose 3-tap patterns are rotated crosses/diagonals
// around (y,x) in the ORIGINAL frame, and whose float4 lanes land in the
// thread's own 2x2 HR block with a per-rotation lane permutation.
// => no rot90, no padding, no atomics; one thread = one 2x2 output block.

#define USE_ASYNC_TILE 1   // gfx1250 global_load_async_to_lds_b32 for halo tile

__global__ __launch_bounds__(256) void hllut_kernel(
    const int* __restrict__ img,      // 4*512*512
    const float* __restrict__ hw,     // 2^24 * 4
    const float* __restrict__ lw,     // 2^24 * 4
    float* __restrict__ out)          // 4*1024*1024
{
    // 16x16 LR tile per block, +2 halo each side (taps reach +/-2)
    __shared__ int tile[20][20];

    const int tx = threadIdx.x & 15;
    const int ty = threadIdx.x >> 4;
    const int bx = blockIdx.x & 31;           // 512/16 = 32 tiles per row
    const int by = (blockIdx.x >> 5) & 31;
    const int b  = blockIdx.x >> 10;          // image 0..3

    const int* __restrict__ imgb = img + b * (512 * 512);

    // ---- stage clamped 20x20 halo tile into LDS ----
    const int oy = by * 16 - 2;
    const int ox = bx * 16 - 2;
    for (int t = threadIdx.x; t < 400; t += 256) {
        const int lyy = t / 20, lxx = t % 20;
        const int gy = min(max(oy + lyy, 0), 511);
        const int gx = min(max(ox + lxx, 0), 511);
#if USE_ASYNC_TILE
        // per-lane async DMA: VDST = LDS byte address, VADDR = 64-bit global addr
        const uint32_t lds_addr = (uint32_t)(uintptr_t)&tile[lyy][lxx];
        const uint64_t gaddr    = (uint64_t)(uintptr_t)(imgb + (gy << 9) + gx);
        asm volatile("global_load_async_to_lds_b32 %0, %1, off"
                     :: "v"(lds_addr), "v"(gaddr) : "memory");
#else
        tile[lyy][lxx] = imgb[(gy << 9) + gx];
#endif
    }
#if USE_ASYNC_TILE
    asm volatile("s_wait_asynccnt 0x0" ::: "memory");
#endif
    __syncthreads();

    const int y  = by * 16 + ty;
    const int x  = bx * 16 + tx;
    const int ly = ty + 2, lx = tx + 2;

    // 13 taps (edge clamping is baked into the halo fill)
    const int c00 = tile[ly][lx];
    const int e1  = tile[ly][lx + 1], e2 = tile[ly][lx + 2];   // right  (h,r0)
    const int s1  = tile[ly + 1][lx], s2 = tile[ly + 2][lx];   // down   (h,r1)
    const int w1  = tile[ly][lx - 1], w2 = tile[ly][lx - 2];   // left   (h,r2)
    const int n1  = tile[ly - 1][lx], n2 = tile[ly - 2][lx];   // up     (h,r3)
    const int se  = tile[ly + 1][lx + 1];                      // l,r0
    const int sw  = tile[ly + 1][lx - 1];                      // l,r1
    const int nw  = tile[ly - 1][lx - 1];                      // l,r2
    const int ne  = tile[ly - 1][lx + 1];                      // l,r3

    // 8 LUT indices (a*65536 + b*256 + c in the rotated frame)
    const int ih0 = c00 * 65536 + e1 * 256 + e2;
    const int ih1 = c00 * 65536 + s1 * 256 + s2;
    const int ih2 = c00 * 65536 + w1 * 256 + w2;
    const int ih3 = c00 * 65536 + n1 * 256 + n2;
    const int il0 = c00 * 65536 + e1 * 256 + se;
    const int il1 = c00 * 65536 + s1 * 256 + sw;
    const int il2 = c00 * 65536 + w1 * 256 + nw;
    const int il3 = c00 * 65536 + n1 * 256 + ne;

    // 8 independent 16B gathers -> global_load_b128, all in flight together
    const float4* __restrict__ hw4 = (const float4*)hw;
    const float4* __restrict__ lw4 = (const float4*)lw;
    const float4 h0 = hw4[ih0], h1 = hw4[ih1], h2 = hw4[ih2], h3 = hw4[ih3];
    const float4 l0 = lw4[il0], l1 = lw4[il1], l2 = lw4[il2], l3 = lw4[il3];

    // lane permutation per rotation r for output sub-pixel (p,q):
    //   r0: s=2p+q   r1: s=2(1-q)+p   r2: s=2(1-p)+(1-q)   r3: s=2q+(1-p)
    float a00 = h0.x + h1.z + h2.w + h3.y + l0.x + l1.z + l2.w + l3.y;
    float a01 = h0.y + h1.x + h2.z + h3.w + l0.y + l1.x + l2.z + l3.w;
    float a10 = h0.z + h1.w + h2.y + h3.x + l0.z + l1.w + l2.y + l3.x;
    float a11 = h0.w + h1.y + h2.x + h3.z + l0.w + l1.y + l2.x + l3.z;

    // out = sum / 2 ; two coalesced 8B stores
    float* __restrict__ outb = out + b * (1024 * 1024);
    const int Y = 2 * y, X = 2 * x;
    float2 r0v; r0v.x = a00 * 0.5f; r0v.y = a01 * 0.5f;
    float2 r1v; r1v.x = a10 * 0.5f; r1v.y = a11 * 0.5f;
    *(float2*)&outb[(size_t)Y * 1024 + X]       = r0v;
    *(float2*)&outb[(size_t)(Y + 1) * 1024 + X] = r1v;
}

extern "C" void kernel_launch(void* const* d_in, const int* in_sizes, int n_in,
                              void* d_out, int out_size, void* d_ws, size_t ws_size,
                              hipStream_t stream) {
    const int*   img = (const int*)d_in[0];
    const float* hw  = (const float*)d_in[1];
    const float* lw  = (const float*)d_in[2];
    float*       out = (float*)d_out;
    // 4 images * (32x32) tiles = 4096 blocks of 256 threads (8 wave32s)
    hipLaunchKernelGGL(hllut_kernel, dim3(4096), dim3(256), 0, stream,
                       img, hw, lw, out);
}